// GNNEncoder_10522669875348
// MI455X (gfx1250) — compile-verified
//
#include <hip/hip_runtime.h>

#define N_NODES 100000
#define N_EDGES 625000
#define DIM     128
#define N_LAYERS 10

typedef __attribute__((ext_vector_type(2))) float v2f;
typedef __attribute__((ext_vector_type(8))) float v8f;

// ---------------- utility: grid-stride zero ----------------
__global__ void zero_f32(float* __restrict__ p, long n) {
    long i = (long)blockIdx.x * blockDim.x + threadIdx.x;
    long s = (long)gridDim.x * blockDim.x;
    for (; i < n; i += s) p[i] = 0.0f;
}

// ---------------- weight pack: interleave k-pairs per column ----------------
// Wp[layer][(k>>1)*DIM + n][0..1] = { W[layer][k&~1][n], W[layer][k|1][n] }
// so a lane's B operand pair (rows k0,k0+1 at column n) is one b64 load.
__global__ void pack_w(const float* __restrict__ W, float* __restrict__ Wp) {
    int i = blockIdx.x * blockDim.x + threadIdx.x;
    if (i >= N_LAYERS * DIM * DIM) return;
    int layer = i >> 14;            // / (DIM*DIM)
    int rem   = i & (DIM * DIM - 1);
    int k     = rem >> 7;           // / DIM
    int n     = rem & (DIM - 1);
    Wp[(size_t)layer * DIM * DIM + (size_t)(((k >> 1) * DIM + n) << 1) + (k & 1)] = W[i];
}

// ---------------- in-degree (once per launch) ----------------
__global__ void degree_kernel(const int* __restrict__ dst, float* __restrict__ deg) {
    int e = blockIdx.x * blockDim.x + threadIdx.x;
    if (e < N_EDGES) atomicAdd(&deg[dst[e]], 1.0f);
}

// ---------------- edge scatter: one wave32 per edge ----------------
__global__ void scatter_kernel(const float* __restrict__ x,
                               const int* __restrict__ src,
                               const int* __restrict__ dst,
                               float* __restrict__ agg) {
    int t = blockIdx.x * blockDim.x + threadIdx.x;
    int e = t >> 5;
    if (e >= N_EDGES) return;
    int lane = t & 31;
    int s = src[e];
    int d = dst[e];
    const float4 v = *(const float4*)(x + (size_t)s * DIM + lane * 4);
    float* o = agg + (size_t)d * DIM + lane * 4;
    atomicAdd(o + 0, v.x);
    atomicAdd(o + 1, v.y);
    atomicAdd(o + 2, v.z);
    atomicAdd(o + 3, v.w);
}

// ---------------- fused SAGE layer GEMM ----------------
// out[16 rows x 128 cols per block] = (agg/deg) @ Wl + x @ Wr + b, optional ReLU.
// 8 waves/block; wave w owns column tile [16w, 16w+16).
// K-loop: 32 steps of V_WMMA_F32_16X16X4_F32 x2 (Wl path, Wr path),
// A from LDS (ds_load_2addr_b64), B from packed weights (global_load_b64).
__global__ void __launch_bounds__(256)
sage_gemm_kernel(const float* __restrict__ agg, const float* __restrict__ xin,
                 const float* __restrict__ deg,
                 const float* __restrict__ WlP, const float* __restrict__ WrP,
                 const float* __restrict__ bias,
                 float* __restrict__ out, int relu) {
    __shared__ float sA[16][DIM + 4];   // mean-aggregated neighbor features
    __shared__ float sX[16][DIM + 4];   // root features

    const int tid = threadIdx.x;
    const int rb  = blockIdx.x;         // 16-row block; N_NODES == 6250*16 exactly

    // cooperative stage: 256 threads x 8 floats = 16x128 per array
    {
        int row = tid >> 4;             // 0..15
        int col = (tid & 15) * 8;       // 0..120
        size_t g = (size_t)(rb * 16 + row) * DIM + col;
        float inv = 1.0f / fmaxf(deg[rb * 16 + row], 1.0f);
        float4 a0 = *(const float4*)(agg + g);
        float4 a1 = *(const float4*)(agg + g + 4);
        float4 x0 = *(const float4*)(xin + g);
        float4 x1 = *(const float4*)(xin + g + 4);
        sA[row][col + 0] = a0.x * inv;  sA[row][col + 1] = a0.y * inv;
        sA[row][col + 2] = a0.z * inv;  sA[row][col + 3] = a0.w * inv;
        sA[row][col + 4] = a1.x * inv;  sA[row][col + 5] = a1.y * inv;
        sA[row][col + 6] = a1.z * inv;  sA[row][col + 7] = a1.w * inv;
        sX[row][col + 0] = x0.x;  sX[row][col + 1] = x0.y;
        sX[row][col + 2] = x0.z;  sX[row][col + 3] = x0.w;
        sX[row][col + 4] = x1.x;  sX[row][col + 5] = x1.y;
        sX[row][col + 6] = x1.z;  sX[row][col + 7] = x1.w;
    }
    __syncthreads();

    const int lane  = tid & 31;
    const int wv    = tid >> 5;          // 0..7 -> column tile
    const int n0    = wv * 16;
    const int nn    = lane & 15;         // A-row index AND B/C column index
    const int khalf = (lane >> 4) << 1;  // 0 or 2: K sub-slot within 16x16x4 op

    // accumulator init = bias (bias depends only on output column)
    v8f c;
    {
        float bv = bias[n0 + nn];
        #pragma unroll
        for (int r = 0; r < 8; ++r) c[r] = bv;
    }

    // packed-B base: rows (khalf, khalf+1) at column (n0+nn) are 8 contiguous bytes
    const float* wlp = WlP + (size_t)khalf * DIM + (size_t)(n0 + nn) * 2;
    const float* wrp = WrP + (size_t)khalf * DIM + (size_t)(n0 + nn) * 2;

    #pragma unroll 4
    for (int kb = 0; kb < DIM; kb += 4) {
        const int k0 = kb + khalf;
        v2f aA = *(const v2f*)&sA[nn][k0];
        v2f aX = *(const v2f*)&sX[nn][k0];
        v2f bL = *(const v2f*)(wlp + (size_t)kb * DIM);
        v2f bR = *(const v2f*)(wrp + (size_t)kb * DIM);
        c = __builtin_amdgcn_wmma_f32_16x16x4_f32(false, aA, false, bL,
                                                  (short)0, c, false, false);
        c = __builtin_amdgcn_wmma_f32_16x16x4_f32(false, aX, false, bR,
                                                  (short)0, c, false, false);
    }

    // C/D layout: VGPR r -> row (r + 8*(lane>>4)), col nn
    size_t outbase = (size_t)(rb * 16) * DIM + n0 + nn;
    const int rofs = (lane >> 4) << 3;
    #pragma unroll
    for (int r = 0; r < 8; ++r) {
        float v = c[r];
        if (relu) v = fmaxf(v, 0.0f);
        out[outbase + (size_t)(r + rofs) * DIM] = v;
    }
}

extern "C" void kernel_launch(void* const* d_in, const int* in_sizes, int n_in,
                              void* d_out, int out_size, void* d_ws, size_t ws_size,
                              hipStream_t stream) {
    const float* x    = (const float*)d_in[0];
    const float* Wl   = (const float*)d_in[1];
    const float* Wr   = (const float*)d_in[2];
    const float* bias = (const float*)d_in[3];
    const int*   edge = (const int*)d_in[4];
    const int*   src  = edge;               // edge_index[0]
    const int*   dst  = edge + N_EDGES;     // edge_index[1]
    float* out = (float*)d_out;

    // workspace: agg (N*D) | ping buffer (N*D) | deg (N) | WlP | WrP
    float* agg  = (float*)d_ws;
    float* buf  = agg + (size_t)N_NODES * DIM;
    float* deg  = buf + (size_t)N_NODES * DIM;
    float* wlP  = deg + (size_t)N_NODES;
    float* wrP  = wlP + (size_t)N_LAYERS * DIM * DIM;

    const int nW = N_LAYERS * DIM * DIM;
    pack_w<<<(nW + 255) / 256, 256, 0, stream>>>(Wl, wlP);
    pack_w<<<(nW + 255) / 256, 256, 0, stream>>>(Wr, wrP);

    // in-degree (deterministic, recomputed each launch)
    zero_f32<<<1024, 256, 0, stream>>>(deg, (long)N_NODES);
    degree_kernel<<<(N_EDGES + 255) / 256, 256, 0, stream>>>(dst, deg);

    const float* cur = x;
    for (int i = 0; i < N_LAYERS; ++i) {
        zero_f32<<<4096, 256, 0, stream>>>(agg, (long)N_NODES * DIM);
        scatter_kernel<<<(N_EDGES * 32 + 255) / 256, 256, 0, stream>>>(cur, src, dst, agg);
        // parity: layer 9 (odd) must land in d_out
        float* nxt = (i & 1) ? out : buf;
        sage_gemm_kernel<<<N_NODES / 16, 256, 0, stream>>>(
            agg, cur, deg,
            wlP + (size_t)i * DIM * DIM, wrP + (size_t)i * DIM * DIM,
            bias + (size_t)i * DIM,
            nxt, (i < N_LAYERS - 1) ? 1 : 0);
        cur = nxt;
    }
}